// LinearShapeFunction_68779606278320
// MI455X (gfx1250) — compile-verified
//
#include <hip/hip_runtime.h>

// LinearShapeFunction: N=2M particles, DIM=3, 8-node trilinear tent basis.
// Bandwidth-bound streaming kernel: 140 B/particle (12 read, 128 written),
// ~0.28 GB/call -> ~12 us floor at 23.3 TB/s. ~100 flops/particle => WMMA
// is not applicable; optimize data movement (b128 NT stores, wave32 blocks).

#define INV_CELL 20.0f

typedef float v4f __attribute__((ext_vector_type(4)));

// Per-axis tent basis for the two surrounding nodes (offset 0 and 1).
// Matches reference exactly: dist = rel - (base + offset);
// basis = |dist|<1 ? 1-|dist| : 0 ; dbasis = |dist|<1 ? -sign(dist)*20 : 0.
__device__ __forceinline__ void axis_basis(float rel,
                                           float& b0, float& b1,
                                           float& g0, float& g1) {
    float base = floorf(rel);
    float d0 = rel - base;            // offset 0
    float d1 = rel - (base + 1.0f);   // offset 1
    float a0 = fabsf(d0);
    float a1 = fabsf(d1);
    bool in0 = a0 < 1.0f;
    bool in1 = a1 < 1.0f;
    b0 = in0 ? (1.0f - a0) : 0.0f;
    b1 = in1 ? (1.0f - a1) : 0.0f;
    float s0 = (d0 > 0.0f) ? 1.0f : ((d0 < 0.0f) ? -1.0f : 0.0f);
    float s1 = (d1 > 0.0f) ? 1.0f : ((d1 < 0.0f) ? -1.0f : 0.0f);
    g0 = in0 ? (-s0 * INV_CELL) : 0.0f;
    g1 = in1 ? (-s1 * INV_CELL) : 0.0f;
}

__global__ __launch_bounds__(256) void
LinearShapeFunction_kernel(const float* __restrict__ pos,
                           float* __restrict__ shapef,   // [N,8]
                           float* __restrict__ grad,     // [N,8,3]
                           int n) {
    int i = blockIdx.x * blockDim.x + (int)threadIdx.x;
    if (i >= n) return;

    const size_t base_in = (size_t)i * 3;

    // Stream-ahead hint for the read side (lowers to global_prefetch_b8).
    // Speculative prefetch: translation failures past the end are dropped.
    __builtin_prefetch(pos + base_in + 3 * 1024, 0, 1);

    float px = pos[base_in + 0];
    float py = pos[base_in + 1];
    float pz = pos[base_in + 2];

    float bx[2], gx[2], by[2], gy[2], bz[2], gz[2];
    axis_basis(px * INV_CELL, bx[0], bx[1], gx[0], gx[1]);
    axis_basis(py * INV_CELL, by[0], by[1], gy[0], gy[1]);
    axis_basis(pz * INV_CELL, bz[0], bz[1], gz[0], gz[1]);

    // OFFSETS row k = ((k>>2)&1, (k>>1)&1, k&1)  [meshgrid 'ij' order]
    float s[8];
    float g[24];
#pragma unroll
    for (int k = 0; k < 8; ++k) {
        const int ox = (k >> 2) & 1;
        const int oy = (k >> 1) & 1;
        const int oz = k & 1;
        const float Bx = bx[ox], By = by[oy], Bz = bz[oz];
        s[k]         = (Bx * By) * Bz;              // prod(basis, axis=-1)
        g[3 * k + 0] = (gx[ox] * By) * Bz;          // dbasis0 * b1 * b2
        g[3 * k + 1] = (gy[oy] * Bx) * Bz;          // dbasis1 * b0 * b2
        g[3 * k + 2] = (gz[oz] * Bx) * By;          // dbasis2 * b0 * b1
    }

    // Write-once outputs: 128-bit non-temporal stores, all 16B-aligned.
    v4f* sp = reinterpret_cast<v4f*>(shapef + (size_t)i * 8);
    v4f sv0 = { s[0], s[1], s[2], s[3] };
    v4f sv1 = { s[4], s[5], s[6], s[7] };
    __builtin_nontemporal_store(sv0, sp + 0);
    __builtin_nontemporal_store(sv1, sp + 1);

    v4f* gp = reinterpret_cast<v4f*>(grad + (size_t)i * 24);
#pragma unroll
    for (int q = 0; q < 6; ++q) {
        v4f gv = { g[4 * q + 0], g[4 * q + 1], g[4 * q + 2], g[4 * q + 3] };
        __builtin_nontemporal_store(gv, gp + q);
    }
}

extern "C" void kernel_launch(void* const* d_in, const int* in_sizes, int n_in,
                              void* d_out, int out_size, void* d_ws, size_t ws_size,
                              hipStream_t stream) {
    const float* pos = (const float*)d_in[0];
    const int n = in_sizes[0] / 3;            // flat count -> particle count

    float* shapef = (float*)d_out;            // [n,8]
    float* grad   = shapef + (size_t)n * 8;   // [n,8,3] concatenated after

    const int block = 256;                    // 8 wave32 waves per block
    const int grid  = (n + block - 1) / block;
    LinearShapeFunction_kernel<<<grid, block, 0, stream>>>(pos, shapef, grad, n);
}